// SphericalGridInterpolator_62216896250097
// MI455X (gfx1250) — compile-verified
//
#include <hip/hip_runtime.h>
#include <hip/hip_bf16.h>
#include <math.h>

// Spherical grid slerp interpolation, MI455X (gfx1250).
//
// Write-bandwidth-bound: 256 MB out + 16 MB in -> ~12 us floor at 23.3 TB/s.
// - 256 KB feature grid staged to LDS (320 KB/WGP) via CDNA5 async
//   global->LDS engine (ASYNCcnt / s_wait_asynccnt); all 4-corner gathers
//   are served by LDS.
// - 4 points per thread: b128 non-temporal output stores (out > L2, so NT
//   avoids thrashing the 192 MB L2) and float4 input loads.

#define FEAT 32
#define NA   64
#define NE   32
#define BLK  512   // 16 wave32s

typedef float v4f __attribute__((ext_vector_type(4)));

__device__ __forceinline__ void async_load_b128(unsigned lds_addr, const void* gaddr) {
  // CDNA5 async memory path: copies 16 bytes from the global address in vaddr to the LDS address in vdst, tracked by ASYNCcnt.
  asm volatile("global_load_async_to_lds_b128 %0, %1, off"
               :: "v"(lds_addr), "v"(gaddr) : "memory");
}

struct PParams {
  float w1a, w2a, w1e, w2e;
  int   c_bl, c_br, c_tl, c_tr;
  bool  south, north;
};

__device__ __forceinline__ void compute_params(float az, float el, PParams& P) {
  const float PI_F      = 3.14159265358979323846f;
  const float OM_AZ     = PI_F / 32.0f;            // 2*pi / NA
  const float OM_EL     = PI_F / 33.0f;            // pi / (NE + 1)
  const float INV_OM_AZ = 32.0f / PI_F;
  const float INV_OM_EL = 33.0f / PI_F;
  const float inv_so_az = 1.0f / sinf(OM_AZ);      // constant-folded
  const float inv_so_el = 1.0f / sinf(OM_EL);

  // searchsorted(ticks_az, az, 'left') on uniform ticks: count of ticks < az
  const float az01 = (az + PI_F) * INV_OM_AZ;      // in [0, 64)
  int ar = (int)ceilf(az01);
  int al = ar - 1;
  if (ar >= NA) ar = 0;                            // azimuth wrap
  if (al < 0)  al = NA - 1;                        // torch negative-index wrap
  const float theta_az = az + PI_F - (float)al * OM_AZ;  // az - ticks_az[al]

  // elevation: interior ticks are -pi/2 + (j+1)*OM_EL, j = 0..NE-1
  const float t = (el + 0.5f * PI_F) * INV_OM_EL - 1.0f;
  int er = (int)ceilf(t);
  er = min(max(er, 0), NE);
  P.south = (er == 0);
  P.north = (er == NE);
  const int eil = min(max(er - 1, 0), NE - 1);
  const int eir = min(er, NE - 1);

  const float base_el = P.south ? (-0.5f * PI_F)
                                : (-0.5f * PI_F + (float)(eil + 1) * OM_EL);
  const float theta_el = el - base_el;

  P.w1a = __sinf(OM_AZ - theta_az) * inv_so_az;
  P.w2a = __sinf(theta_az) * inv_so_az;
  P.w1e = __sinf(OM_EL - theta_el) * inv_so_el;
  P.w2e = __sinf(theta_el) * inv_so_el;

  P.c_bl = al * NE + eil;
  P.c_br = ar * NE + eil;
  P.c_tl = al * NE + eir;
  P.c_tr = ar * NE + eir;
}

__global__ __launch_bounds__(BLK)
void SphericalGridInterpolator_kernel(const float* __restrict__ pts,   // [2, N]
                                      const float* __restrict__ grid,  // [FEAT, NA, NE]
                                      const float* __restrict__ poles, // [FEAT, 2]
                                      float* __restrict__ out,         // [FEAT, N]
                                      int n)
{
  __shared__ __align__(16) float s_grid[FEAT * NA * NE]; // 256 KB
  __shared__ __align__(16) float s_pole[FEAT * 2];       // 256 B

  const int tid = threadIdx.x;

  // ---- Stage grid + poles into LDS via async global->LDS (B128) ----
  for (int v = tid; v < (FEAT * NA * NE) / 4; v += BLK) {
    async_load_b128((unsigned)(uintptr_t)&s_grid[v * 4],
                    (const void*)(grid + (size_t)v * 4));
  }
  if (tid < (FEAT * 2) / 4) {
    async_load_b128((unsigned)(uintptr_t)&s_pole[tid * 4],
                    (const void*)(poles + (size_t)tid * 4));
  }
  asm volatile("s_wait_asynccnt 0" ::: "memory"); // drain this wave's async ops
  __syncthreads();                                 // make all staging visible

  const int gtid   = blockIdx.x * blockDim.x + tid;
  const int stride = gridDim.x * blockDim.x;

  if ((n & 3) == 0) {
    // ---- Vector path: 4 consecutive points per thread, b128 NT stores ----
    const int n4 = n >> 2;
    for (int g = gtid; g < n4; g += stride) {
      const int i = g << 2;
      const v4f az4 = *(const v4f*)(pts + i);
      const v4f el4 = *(const v4f*)(pts + n + i);

      PParams P[4];
#pragma unroll
      for (int p = 0; p < 4; ++p) compute_params(az4[p], el4[p], P[p]);

#pragma unroll
      for (int f = 0; f < FEAT; ++f) {
        const float* gf = s_grid + f * (NA * NE);
        v4f r;
#pragma unroll
        for (int p = 0; p < 4; ++p) {
          const float fb = P[p].w1a * gf[P[p].c_bl] + P[p].w2a * gf[P[p].c_br];
          const float ft = P[p].w1a * gf[P[p].c_tl] + P[p].w2a * gf[P[p].c_tr];
          const float f1 = P[p].south ? s_pole[2 * f]     : fb;
          const float f2 = P[p].north ? s_pole[2 * f + 1] : ft;
          r[p] = P[p].w1e * f1 + P[p].w2e * f2;
        }
        // streaming 16B store, non-temporal (output > L2 capacity)
        __builtin_nontemporal_store(r, (v4f*)(out + (size_t)f * (size_t)n + i));
      }
    }
  } else {
    // ---- Scalar fallback (n not a multiple of 4) ----
    for (int i = gtid; i < n; i += stride) {
      PParams P;
      compute_params(pts[i], pts[n + i], P);
#pragma unroll
      for (int f = 0; f < FEAT; ++f) {
        const float* gf = s_grid + f * (NA * NE);
        const float fb = P.w1a * gf[P.c_bl] + P.w2a * gf[P.c_br];
        const float ft = P.w1a * gf[P.c_tl] + P.w2a * gf[P.c_tr];
        const float f1 = P.south ? s_pole[2 * f]     : fb;
        const float f2 = P.north ? s_pole[2 * f + 1] : ft;
        __builtin_nontemporal_store(P.w1e * f1 + P.w2e * f2,
                                    out + (size_t)f * (size_t)n + i);
      }
    }
  }
}

extern "C" void kernel_launch(void* const* d_in, const int* in_sizes, int n_in,
                              void* d_out, int out_size, void* d_ws, size_t ws_size,
                              hipStream_t stream) {
  const float* pts   = (const float*)d_in[0];  // (2, N)
  const float* grid  = (const float*)d_in[1];  // (FEAT, NA, NE)
  const float* poles = (const float*)d_in[2];  // (FEAT, 2)
  float* out = (float*)d_out;                  // (FEAT, N)
  const int n = in_sizes[0] / 2;

  dim3 gdim(1024), bdim(BLK);
  SphericalGridInterpolator_kernel<<<gdim, bdim, 0, stream>>>(pts, grid, poles, out, n);
}